// HowlingTransform_13125420056586
// MI455X (gfx1250) — compile-verified
//
#include <hip/hip_runtime.h>
#include <math.h>

typedef __attribute__((ext_vector_type(2))) float v2f;
typedef __attribute__((ext_vector_type(4))) float v4f;
typedef __attribute__((ext_vector_type(8))) float v8f;

#define SAMPLE_LEN 65536
#define FRAME_LEN  128
#define HOP        64
#define IR_LEN     1024
#define CONV_LEN   1151          // FRAME_LEN + IR_LEN - 1
#define NFB        1006          // feedback iterations; total frames = 1007
#define NT         256           // 8 wave32 waves on one WGP

__device__ __forceinline__ float clamp1(float v) {
  return fminf(1.0f, fmaxf(-1.0f, v));
}

// NTILE independent 16-output conv tiles, each a 32-deep f32 WMMA K-chain.
// A[m][k] = reversed-IR slice (ascending-pair ds loads, no swaps),
// B[k][n] = w[4c+k] broadcast. Writeback fuses the reference's add-then-clip:
// each conv output is owned by exactly one writer lane, once per iteration.
template <int NTILE>
__device__ __forceinline__ void conv_group(float* xs, const float* irr,
                                           const v2f* bArr, int fs64,
                                           int tileBase, int qBase, int mOff,
                                           bool writer) {
  v8f d[NTILE] = {};
#pragma unroll
  for (int c = 0; c < 32; ++c) {
    const v2f bm = bArr[c];
#pragma unroll
    for (int u = 0; u < NTILE; ++u) {
      const int q = qBase - tileBase - 16 * u + 4 * c;
      v2f a;
      a.x = irr[q];                 // ascending pair -> single ds_load_2addr
      a.y = irr[q + 1];
      d[u] = __builtin_amdgcn_wmma_f32_16x16x4_f32(false, a, false, bm,
                                                   (short)0, d[u], false, false);
    }
  }
  if (writer) {                     // lanes 0 (M=0..7) and 16 (M=8..15)
#pragma unroll
    for (int u = 0; u < NTILE; ++u) {
      float* p = &xs[fs64 + tileBase + 16 * u + mOff];  // 32B-aligned
      v4f lo = *(const v4f*)p;
      v4f hi = *(const v4f*)(p + 4);
      lo.x = clamp1(lo.x + d[u][0]); lo.y = clamp1(lo.y + d[u][1]);
      lo.z = clamp1(lo.z + d[u][2]); lo.w = clamp1(lo.w + d[u][3]);
      hi.x = clamp1(hi.x + d[u][4]); hi.y = clamp1(hi.y + d[u][5]);
      hi.z = clamp1(hi.z + d[u][6]); hi.w = clamp1(hi.w + d[u][7]);
      *(v4f*)p = lo;        // tile 71 e=7: clamp(x + 0.0) == reference clip
      *(v4f*)(p + 4) = hi;
    }
  }
}

__global__ __launch_bounds__(NT, 1)
void howling_kernel(const float* __restrict__ x_in,
                    const float* __restrict__ ir_in,
                    float* __restrict__ out) {
  // ---- LDS partition (dynamic: ~276 KB, fits CDNA5's 320 KB/WGP) ----
  extern __shared__ float smem[];
  float* xs   = smem;             // 65536 : full signal, resident for whole sim
  float* irp  = xs + SAMPLE_LEN;  // 1280  : raw padded IR (DFT input)
  float* irr  = irp + 1280;       // 1280  : scaled, REVERSED padded IR
  float* wbuf = irr + 1280;       // 128   : current windowed frame
  float* accb = wbuf + 128;       // 128   : overlap-add rolling carry
  float* winb = accb + 128;       // 128   : periodic Hann window
  float* red  = winb + 128;       // 520   : DFT masked powers + scale scalar

  const int tid  = threadIdx.x;
  const int lane = tid & 31;
  const int wave = tid >> 5;

  // ---- init: load x, window, zero IR pad / carry ----
  for (int j = tid; j < SAMPLE_LEN; j += NT) xs[j] = x_in[j];
  if (tid < FRAME_LEN) {
    winb[tid] = 0.5f * (1.0f - __cosf(6.283185307179586f * (float)tid / 128.0f));
    accb[tid] = 0.0f;
  }
  for (int j = tid; j < 1280; j += NT) irp[j] = 0.0f;
  __syncthreads();
  for (int j = tid; j < IR_LEN; j += NT) irp[128 + j] = ir_in[j];
  __syncthreads();

  // ---- IR scaling: 513-bin direct DFT, exact (k*t mod 1024) phase ----
  for (int k = tid; k <= 512; k += NT) {
    float re = 0.0f, im = 0.0f;
    for (int t = 0; t < IR_LEN; ++t) {
      int   ph  = (k * t) & 1023;
      float ang = (float)ph * (-6.283185307179586f / 1024.0f);
      float s, c;
      __sincosf(ang, &s, &c);
      float v = irp[128 + t];
      re = fmaf(v, c, re);
      im = fmaf(v, s, im);
    }
    float p     = re * re + im * im;
    float phase = atan2f(im, re);
    red[k] = (phase > -0.1f && phase < 0.1f) ? p : -3.0e38f;
  }
  __syncthreads();
  if (tid == 0) {
    float peak = -3.0e38f;
    for (int k = 0; k <= 512; ++k) peak = fmaxf(peak, red[k]);
    // ir/factor == ir * 2^((2 - 10*log10(peak))/6)  (MLG cancels)
    red[516] = exp2f((2.0f - 10.0f * log10f(peak)) / 6.0f);
  }
  __syncthreads();
  {
    float s = red[516];
    for (int j = tid; j < 1280; j += NT) irr[j] = irp[1279 - j] * s;
  }
  __syncthreads();

  // WMMA lane-role constants (per CDNA5 ISA §7.12.2 f32 layouts)
  const int  kSel   = (lane >> 4) << 1;   // A: VGPR0 K=0|2, VGPR1 K=1|3; B rows same split
  const int  mRow   = lane & 15;          // A: M = lane % 16
  const int  mOff   = (lane >> 4) << 3;   // D: lanes>=16 hold M = v+8
  const int  qBase  = 1151 + kSel - mRow; // reversed-IR index base
  const bool writer = (lane & 15) == 0;   // lanes 0 / 16 own full D half-rows

  // ---- serial feedback loop: whole simulation on one WGP ----
  for (int i = 0; i <= NFB; ++i) {
    const int fs = i * HOP;

    // windowed frame + overlap-add carry
    if (tid < FRAME_LEN) {
      float wv = winb[tid] * xs[fs + tid];
      wbuf[tid] = wv;
      accb[tid] += wv;
    }
    __syncthreads();

    // low half of carry is final: emit clipped howl samples, shift carry
    if (tid < HOP) {
      out[fs + tid] = clamp1(accb[tid]);
      float hi = accb[tid + HOP];
      accb[tid]       = hi;
      accb[tid + HOP] = 0.0f;
    }

    if (i < NFB) {
      const int fs64 = fs + HOP;
      // B pairs (w broadcast); compiler folds into immediate-offset ds loads
      v2f bArr[32];
#pragma unroll
      for (int c = 0; c < 32; ++c) {
        bArr[c].x = wbuf[4 * c + kSel];
        bArr[c].y = wbuf[4 * c + kSel + 1];
      }
      // 9 tiles per wave as 4+4+1 independent WMMA chains (72 tiles total),
      // writeback applies x = clip(x + conv) directly.
      const int tb = wave * 144;
      conv_group<4>(xs, irr, bArr, fs64, tb,       qBase, mOff, writer);
      conv_group<4>(xs, irr, bArr, fs64, tb + 64,  qBase, mOff, writer);
      conv_group<1>(xs, irr, bArr, fs64, tb + 128, qBase, mOff, writer);
    }

    if (i == 0) {
      // reference clips the ENTIRE (initially unclipped) x on iteration 0;
      // conv region is already clamped by the fused writeback (idempotent).
      __syncthreads();
      for (int j = tid; j < SAMPLE_LEN; j += NT) xs[j] = clamp1(xs[j]);
    }
    __syncthreads();
  }

  // flush remaining carry + zero tail of howl output
  if (tid < HOP) out[64448 + tid] = clamp1(accb[tid]);
  for (int j = 64512 + tid; j < SAMPLE_LEN; j += NT) out[j] = 0.0f;
}

extern "C" void kernel_launch(void* const* d_in, const int* in_sizes, int n_in,
                              void* d_out, int out_size, void* d_ws, size_t ws_size,
                              hipStream_t stream) {
  const float* x  = (const float*)d_in[0];
  const float* ir = (const float*)d_in[1];
  float* outp = (float*)d_out;
  size_t smem_bytes =
      (size_t)(SAMPLE_LEN + 1280 + 1280 + 128 + 128 + 128 + 520) * sizeof(float);
  howling_kernel<<<1, NT, smem_bytes, stream>>>(x, ir, outp);
}